// KrauseAttention_49143015801153
// MI455X (gfx1250) — compile-verified
//
#include <hip/hip_runtime.h>

typedef __attribute__((ext_vector_type(2))) float v2f;
typedef __attribute__((ext_vector_type(4))) float v4f;
typedef __attribute__((ext_vector_type(8))) float v8f;

#define Bsz  128
#define Hn   32
#define Dd   128
#define Wsz  128
#define TOPK 96
#define Ee   4096   // H*D

// ---------------------------------------------------------------------------
// GEMM: out[m,n] = sum_k A[m,k] * W[n,k] + bias[n]   (y = A @ W^T + b)
// M=128, N=4096, K=4096.  One wave computes a 16x16 tile with
// V_WMMA_F32_16X16X4_F32, marching K in steps of 4 (16x unrolled for a deep
// VMEM pipeline; ~2048 waves keep >12MB of loads in flight to saturate HBM).
// grid = (N/64, M/16), block = 128 threads (4 waves, 16 cols each).
// ---------------------------------------------------------------------------
__global__ __launch_bounds__(128) void gemm_nt_wmma(
    const float* __restrict__ A, const float* __restrict__ W,
    const float* __restrict__ bias, float* __restrict__ out)
{
  const int K = Ee, N = Ee;
  const int lane = threadIdx.x & 31;
  const int wave = threadIdx.x >> 5;
  const int n0   = blockIdx.x * 64 + wave * 16;
  const int m0   = blockIdx.y * 16;
  const int half = lane >> 4;     // 0: K pair {0,1}, 1: K pair {2,3}
  const int r16  = lane & 15;

  const float* aPtr = A + (size_t)(m0 + r16) * K + 2 * half;
  const float* wPtr = W + (size_t)(n0 + r16) * K + 2 * half;

  v8f acc = {};
#pragma unroll 16
  for (int kk = 0; kk < K; kk += 4) {
    v2f a = *(const v2f*)(aPtr + kk);
    v2f b = *(const v2f*)(wPtr + kk);
    // D = A(16x4) x B(4x16) + C, fp32
    acc = __builtin_amdgcn_wmma_f32_16x16x4_f32(
        /*neg_a=*/false, a, /*neg_b=*/false, b,
        /*c_mod=*/(short)0, acc, /*reuse_a=*/false, /*reuse_b=*/false);
  }

  const int   n  = n0 + r16;
  const float bn = bias[n];
#pragma unroll
  for (int r = 0; r < 8; ++r) {
    const int m = m0 + r + 8 * half;   // C/D layout: lanes 16-31 hold M=8..15
    out[(size_t)m * N + n] = acc[r] + bn;
  }
}

// ---------------------------------------------------------------------------
// Ring update: out = ring with slot (pos % W) replaced by newv[b,h,:].
// Pure streaming copy (read-once/write-once, ~1GB total): use non-temporal
// hints (TH=NT) so this stream does not evict the L2-resident weight
// matrices / x / Q that the GEMM and attention kernels reuse.
// v4f-vectorized; D=128 divisible by 4 so a vector never straddles rows.
// ---------------------------------------------------------------------------
__global__ __launch_bounds__(256) void ring_update(
    const float* __restrict__ ring, const float* __restrict__ newv,
    const int* __restrict__ pos, float* __restrict__ out)
{
  const int slot = pos[0] & (Wsz - 1);                 // W is a power of two
  const long long i4 = (long long)blockIdx.x * blockDim.x + threadIdx.x;
  const long long e  = i4 * 4;                         // element index in [B,H,W,D]
  const int w = (int)((e >> 7) & (Wsz - 1));           // (e / D) % W
  v4f val;
  if (w == slot) {
    const long long bh = e >> 14;                      // e / (W*D)
    const int d = (int)(e & (Dd - 1));
    val = *(const v4f*)(newv + bh * Dd + d);
  } else {
    val = __builtin_nontemporal_load((const v4f*)(ring + e));
  }
  __builtin_nontemporal_store(val, (v4f*)(out + e));
}

// ---------------------------------------------------------------------------
// Gaussian-RBF top-k attention. One 128-thread block (4 waves) per (b,h).
// Roll-invariant: top-k + softmax + gather don't depend on window order.
// K_new/V_new are each read exactly once -> NT loads.
// ---------------------------------------------------------------------------
__global__ __launch_bounds__(128) void rbf_topk_attn(
    const float* __restrict__ Q,    // [B,E]  (head h occupies cols h*D..)
    const float* __restrict__ Kw,   // [B,H,W,D]  (updated ring = K_new)
    const float* __restrict__ Vw,   // [B,H,W,D]  (updated ring = V_new)
    const float* __restrict__ lsig, // [H]
    float* __restrict__ consensus)  // [B,E]
{
  __shared__ __align__(16) float q_s[Dd];
  __shared__ float sc[Wsz];
  __shared__ float pr[Wsz];

  const int b = blockIdx.x / Hn;
  const int h = blockIdx.x % Hn;
  const int tid  = threadIdx.x;
  const int lane = tid & 31;
  const int wave = tid >> 5;
  const size_t baseKV = ((size_t)(b * Hn + h)) * Wsz * Dd;

  q_s[tid] = Q[(size_t)b * Ee + h * Dd + tid];
  __syncthreads();

  const float inv = -0.5f / __expf(2.0f * lsig[h]);    // 1/(-2*sigma^2)

  // squared distances: each wave covers rows {wave, wave+4, ...},
  // 32 lanes split D=128 as v4f each, then wave-reduce.
  for (int r = wave; r < Wsz; r += 4) {
    v4f kv = __builtin_nontemporal_load(
        (const v4f*)(Kw + baseKV + (size_t)r * Dd + lane * 4));
    v4f qv = *(const v4f*)(&q_s[lane * 4]);
    float d0 = qv.x - kv.x, d1 = qv.y - kv.y;
    float d2 = qv.z - kv.z, d3 = qv.w - kv.w;
    float p = d0 * d0 + d1 * d1 + d2 * d2 + d3 * d3;
    p += __shfl_xor(p, 16);
    p += __shfl_xor(p, 8);
    p += __shfl_xor(p, 4);
    p += __shfl_xor(p, 2);
    p += __shfl_xor(p, 1);
    if (lane == 0) sc[r] = p * inv;
  }
  __syncthreads();

  // rank (with jax top_k's lower-index tie-break) + global max
  const float my = sc[tid];
  int rank = 0;
  float mx = -3.402823466e+38f;
#pragma unroll 4
  for (int j = 0; j < Wsz; ++j) {
    const float s = sc[j];
    rank += (s > my || (s == my && j < tid)) ? 1 : 0;
    mx = fmaxf(mx, s);
  }
  // selected set = top-96; its max is the global max (rank-0 always selected)
  const float ex = (rank < TOPK) ? __expf(my - mx) : 0.0f;
  pr[tid] = ex;
  __syncthreads();

  float sum = 0.0f;
#pragma unroll 4
  for (int j = 0; j < Wsz; ++j) sum += pr[j];
  const float pnorm = ex / sum;
  __syncthreads();
  pr[tid] = pnorm;
  __syncthreads();

  // consensus[d] = sum_w p[w] * V[w,d]; lane-contiguous V reads (coalesced)
  float accv = 0.0f;
  for (int w = 0; w < Wsz; ++w) {
    accv += pr[w] * __builtin_nontemporal_load(Vw + baseKV + (size_t)w * Dd + tid);
  }
  consensus[(size_t)b * Ee + h * Dd + tid] = accv;   // RT store: re-read 256x by Wo GEMM
}

// ---------------------------------------------------------------------------
extern "C" void kernel_launch(void* const* d_in, const int* in_sizes, int n_in,
                              void* d_out, int out_size, void* d_ws, size_t ws_size,
                              hipStream_t stream) {
  const float* x    = (const float*)d_in[0];
  const float* Wq   = (const float*)d_in[1];
  const float* bq   = (const float*)d_in[2];
  const float* Wk   = (const float*)d_in[3];
  const float* bk   = (const float*)d_in[4];
  const float* Wv   = (const float*)d_in[5];
  const float* bv   = (const float*)d_in[6];
  const float* Wo   = (const float*)d_in[7];
  const float* bo   = (const float*)d_in[8];
  const float* lsig = (const float*)d_in[9];
  const float* Kring = (const float*)d_in[10];
  const float* Vring = (const float*)d_in[11];
  const int*   pos   = (const int*)d_in[12];

  // d_out = [ out (B*E) | K_new (B*H*W*D) | V_new (B*H*W*D) ]
  float* out  = (float*)d_out;
  float* Knew = out + (size_t)Bsz * Ee;
  float* Vnew = Knew + (size_t)Bsz * Hn * Wsz * Dd;

  // workspace: Q, K, V, consensus, each [B,E] f32 (2 MB each)
  float* Qw = (float*)d_ws;
  float* Kv = Qw + (size_t)Bsz * Ee;
  float* Vv = Kv + (size_t)Bsz * Ee;
  float* Cw = Vv + (size_t)Bsz * Ee;

  dim3 gGemm(Ee / 64, Bsz / 16);   // (64, 8)
  gemm_nt_wmma<<<gGemm, 128, 0, stream>>>(x, Wq, bq, Qw);
  gemm_nt_wmma<<<gGemm, 128, 0, stream>>>(x, Wk, bk, Kv);
  gemm_nt_wmma<<<gGemm, 128, 0, stream>>>(x, Wv, bv, Vv);

  const long long total = (long long)Bsz * Hn * Wsz * Dd;   // 67,108,864
  const int rblocks = (int)(total / 4 / 256);               // 65,536
  ring_update<<<rblocks, 256, 0, stream>>>(Kring, Kv, pos, Knew);
  ring_update<<<rblocks, 256, 0, stream>>>(Vring, Vv, pos, Vnew);

  rbf_topk_attn<<<Bsz * Hn, 128, 0, stream>>>(Qw, Knew, Vnew, lsig, Cw);

  gemm_nt_wmma<<<gGemm, 128, 0, stream>>>(Cw, Wo, bo, out);
}